// Module_27805618274836
// MI455X (gfx1250) — compile-verified
//
#include <hip/hip_runtime.h>
#include <hip/hip_bf16.h>
#include <math.h>

typedef float v2f __attribute__((ext_vector_type(2)));
typedef float v8f __attribute__((ext_vector_type(8)));

// Problem constants (match reference)
#define BB   256
#define LQ   50
#define LK   200
#define DIM  128
#define MT   4      // ceil(50/16)
#define NT1  13     // ceil(200/16)
#define NT2  8      // 128/16

// LDS layout (floats). Strides chosen for bank-conflict-free column reads:
//  stride 132: 132 mod 64 = 4  -> 16 consecutive rows hit 16 distinct banks
//  stride 212: 212 mod 64 = 20 -> 16 consecutive rows hit 16 distinct banks
#define QS_STRIDE 132   // q_s: 64 x 132 (also reused for context output)
#define KV_STRIDE 132   // kv_s: 208 x 132 (k, then reused for v)
#define WS_STRIDE 212   // w_s: 64 x 212 (logits -> weights)
#define QS_ELEMS  (64 * QS_STRIDE)
#define KV_ELEMS  (208 * KV_STRIDE)
#define WS_ELEMS  (64 * WS_STRIDE)
#define SMEM_BYTES ((QS_ELEMS + KV_ELEMS + WS_ELEMS) * 4)

// ---- CDNA5 async global->LDS copy (ASYNCcnt-tracked, fire-and-forget) ----
__device__ __forceinline__ void async_ld_b128(unsigned lds_byte_off, const void* gaddr) {
    // GV mode: each lane copies 16 bytes from the global address in gaddr to the LDS byte offset in lds_byte_off
    asm volatile("global_load_async_to_lds_b128 %0, %1, off"
                 :: "v"(lds_byte_off), "v"(gaddr)
                 : "memory");
}
__device__ __forceinline__ void wait_async0() {
    asm volatile("s_wait_asynccnt 0" ::: "memory");
}

__global__ __launch_bounds__(256) void fused_lognormal_attn(
    const int* __restrict__ Qidx,   // [B, LQ]
    const int* __restrict__ Kidx,   // [B, LK]
    const int* __restrict__ Vidx,   // [B, LK]
    const float* __restrict__ Wq,   // [Nq, D]
    const float* __restrict__ Wk,   // [Nv, D]
    const float* __restrict__ Wv,   // [Nv, D]
    const float* __restrict__ gamma,// [D]
    const float* __restrict__ beta, // [D]
    const float* __restrict__ eps,  // [B, LQ, LK]
    float* __restrict__ out)        // [B, LQ, D]
{
    extern __shared__ float smem[];
    float* q_s  = smem;                 // 64 x QS_STRIDE   (q; later: context)
    float* kv_s = q_s + QS_ELEMS;       // 208 x KV_STRIDE  (k; later: v)
    float* w_s  = kv_s + KV_ELEMS;      // 64 x WS_STRIDE   (logits -> weights)

    const int b    = blockIdx.x;
    const int tid  = threadIdx.x;
    const int lane = tid & 31;
    const int wave = tid >> 5;

    // LDS byte offsets (low 32 bits of generic shared pointer == LDS address)
    const unsigned qs_off = (unsigned)(uintptr_t)q_s;
    const unsigned kv_off = (unsigned)(uintptr_t)kv_s;

    // ---- Phase 0: async-gather q and k rows into LDS, zero padding rows --
    // One wave-iteration copies a full 512B row: lane -> 16B chunk.
    for (int i = tid; i < LQ * (DIM / 4); i += 256) {
        int r  = i >> 5;
        int c4 = (i & 31) << 2;
        async_ld_b128(qs_off + (unsigned)(r * QS_STRIDE + c4) * 4u,
                      Wq + (size_t)Qidx[b * LQ + r] * DIM + c4);
    }
    for (int i = tid; i < LK * (DIM / 4); i += 256) {
        int r  = i >> 5;
        int c4 = (i & 31) << 2;
        async_ld_b128(kv_off + (unsigned)(r * KV_STRIDE + c4) * 4u,
                      Wk + (size_t)Kidx[b * LK + r] * DIM + c4);
    }
    // zero q rows 50..63 (14 x 132) and kv rows 200..207 (8 x 132) via ds
    for (int i = tid; i < 14 * QS_STRIDE; i += 256)
        q_s[50 * QS_STRIDE + i] = 0.0f;
    for (int i = tid; i < 8 * KV_STRIDE; i += 256)
        kv_s[200 * KV_STRIDE + i] = 0.0f;
    wait_async0();
    __syncthreads();

    // ---- Phase 1: logits = q * k^T / sqrt(D) via WMMA f32 16x16x4 -------
    // A 16x4 layout: lane L (M=L&15) holds {A[M][koff], A[M][koff+1]}, koff=(L>>4)*2
    // B 4x16 layout: lane L (N=L&15) holds {B[koff][N], B[koff+1][N]}
    {
        const float scale = 0.088388347648318447f; // 1/sqrt(128)
        const int m16  = lane & 15;
        const int koff = (lane >> 4) << 1;
        for (int t = wave; t < MT * NT1; t += 8) {
            const int mb = (t / NT1) * 16;
            const int nb = (t % NT1) * 16;
            v8f acc = {};
            const float* ap = q_s  + (mb + m16) * QS_STRIDE + koff;
            const float* bp = kv_s + (nb + m16) * KV_STRIDE + koff; // B[K][n] = k[n][K]
            for (int d = 0; d < DIM; d += 4) {
                v2f a = *(const v2f*)(ap + d);
                v2f bfrag = *(const v2f*)(bp + d);
                acc = __builtin_amdgcn_wmma_f32_16x16x4_f32(
                    false, a, false, bfrag, (short)0, acc, false, false);
            }
            // C/D layout: vgpr i, lanes0-15 -> M=i, lanes16-31 -> M=8+i
            const int crow = mb + ((lane >> 4) << 3);
            const int ccol = nb + m16;
#pragma unroll
            for (int i = 0; i < 8; ++i)
                w_s[(crow + i) * WS_STRIDE + ccol] = acc[i] * scale;
        }
    }
    __syncthreads();

    // ---- Phase 2a: async-gather v into kv_s (k dead); overlaps softmax --
    for (int i = tid; i < LK * (DIM / 4); i += 256) {
        int r  = i >> 5;
        int c4 = (i & 31) << 2;
        async_ld_b128(kv_off + (unsigned)(r * KV_STRIDE + c4) * 4u,
                      Wv + (size_t)Vidx[b * LK + r] * DIM + c4);
    }

    // ---- Phase 2b: log-softmax + LogNormal reparam + renormalize --------
    // score = exp(log_phi - 0.5*sigma^2 + sigma*eps), sigma = 1
    for (int r = wave; r < LQ; r += 8) {
        float* row = w_s + r * WS_STRIDE;
        float m = -INFINITY;
        for (int c = lane; c < LK; c += 32) m = fmaxf(m, row[c]);
        for (int o = 16; o; o >>= 1) m = fmaxf(m, __shfl_xor(m, o, 32));
        float s = 0.0f;
        for (int c = lane; c < LK; c += 32) s += __expf(row[c] - m);
        for (int o = 16; o; o >>= 1) s += __shfl_xor(s, o, 32);
        const float logZ = m + __logf(s);
        const float* erow = eps + ((size_t)b * LQ + r) * LK;
        float sc[7];
        float ssum = 0.0f;
        int j = 0;
        for (int c = lane; c < LK; c += 32, ++j) {
            float v = __expf(row[c] - logZ - 0.5f + erow[c]);
            sc[j] = v;
            ssum += v;
        }
        for (int o = 16; o; o >>= 1) ssum += __shfl_xor(ssum, o, 32);
        const float inv = 1.0f / ssum;
        j = 0;
        for (int c = lane; c < LK; c += 32, ++j) row[c] = sc[j] * inv;
    }
    wait_async0();   // drain v copy before it becomes visible post-barrier
    __syncthreads();

    // ---- Phase 3: context = weight * v via WMMA, into q_s region --------
    {
        const int m16  = lane & 15;
        const int koff = (lane >> 4) << 1;
        for (int t = wave; t < MT * NT2; t += 8) {
            const int mb = (t >> 3) * 16;
            const int db = (t & 7) * 16;
            v8f acc = {};
            const float* ap = w_s + (mb + m16) * WS_STRIDE + koff;
            const float* bp = kv_s + koff * KV_STRIDE + db + m16;   // B[K][n] = v[K][n]
            for (int kk = 0; kk < LK; kk += 4) {
                v2f a = *(const v2f*)(ap + kk);
                v2f bfrag;
                bfrag.x = bp[(kk + 0) * KV_STRIDE];
                bfrag.y = bp[(kk + 1) * KV_STRIDE];
                acc = __builtin_amdgcn_wmma_f32_16x16x4_f32(
                    false, a, false, bfrag, (short)0, acc, false, false);
            }
            const int crow = mb + ((lane >> 4) << 3);
            const int ccol = db + m16;
#pragma unroll
            for (int i = 0; i < 8; ++i)
                q_s[(crow + i) * QS_STRIDE + ccol] = acc[i];
        }
    }
    __syncthreads();

    // ---- Phase 4: LayerNorm over D=128 + affine, coalesced store --------
    for (int r = wave; r < LQ; r += 8) {
        const float* row = q_s + r * QS_STRIDE;
        float x[4], sum = 0.0f;
#pragma unroll
        for (int j = 0; j < 4; ++j) { x[j] = row[lane + 32 * j]; sum += x[j]; }
        for (int o = 16; o; o >>= 1) sum += __shfl_xor(sum, o, 32);
        const float mu = sum * (1.0f / 128.0f);
        float vs = 0.0f;
#pragma unroll
        for (int j = 0; j < 4; ++j) { float d = x[j] - mu; vs += d * d; }
        for (int o = 16; o; o >>= 1) vs += __shfl_xor(vs, o, 32);
        const float rstd = rsqrtf(vs * (1.0f / 128.0f) + 1e-5f);
        float* orow = out + ((size_t)b * LQ + r) * DIM;
#pragma unroll
        for (int j = 0; j < 4; ++j) {
            int c = lane + 32 * j;
            orow[c] = (x[j] - mu) * rstd * gamma[c] + beta[c];
        }
    }
}

extern "C" void kernel_launch(void* const* d_in, const int* in_sizes, int n_in,
                              void* d_out, int out_size, void* d_ws, size_t ws_size,
                              hipStream_t stream) {
    const int*   Qidx  = (const int*)d_in[0];
    const int*   Kidx  = (const int*)d_in[1];
    const int*   Vidx  = (const int*)d_in[2];
    const float* Wq    = (const float*)d_in[3];
    const float* Wk    = (const float*)d_in[4];
    const float* Wv    = (const float*)d_in[5];
    const float* gamma = (const float*)d_in[6];
    const float* beta  = (const float*)d_in[7];
    const float* eps   = (const float*)d_in[8];
    float*       out   = (float*)d_out;

    // Allow >64KB dynamic LDS (gfx1250: up to 320KB/WG). Deterministic,
    // not a stream op -> graph-capture safe.
    (void)hipFuncSetAttribute((const void*)fused_lognormal_attn,
                              hipFuncAttributeMaxDynamicSharedMemorySize,
                              SMEM_BYTES);

    fused_lognormal_attn<<<dim3(BB), dim3(256), SMEM_BYTES, stream>>>(
        Qidx, Kidx, Vidx, Wq, Wk, Wv, gamma, beta, eps, out);
}